// RelationalGraphLayer_1889785610906
// MI455X (gfx1250) — compile-verified
//
#include <hip/hip_runtime.h>
#include <cstdint>
#include <cstddef>

typedef __attribute__((ext_vector_type(2))) float v2f;
typedef __attribute__((ext_vector_type(8))) float v8f;

#define DIMC 512
#define NRELC 4

#define BM 128
#define BN 128
#define BK 32
#define LDA_PAD 36   // floats per A row in LDS (144B: 16B aligned, conflict-free)
#define LDB_PAD 136  // floats per B row in LDS (544B: 16B aligned, conflict-free)

// ---------------- CDNA5 async global->LDS copy (ASYNCcnt tracked) ----------
__device__ __forceinline__ void async_copy_b128(uint32_t lds_byte_off, const float* gaddr) {
    asm volatile("global_load_async_to_lds_b128 %0, %1, off"
                 :: "v"(lds_byte_off), "v"(gaddr) : "memory");
}
__device__ __forceinline__ void wait_async_le8() {
    asm volatile("s_wait_asynccnt 0x8" ::: "memory");
}
__device__ __forceinline__ void wait_async_0() {
    asm volatile("s_wait_asynccnt 0x0" ::: "memory");
}

// ---------------- zero fill ----------------
__global__ void zero_kernel(float* __restrict__ p, size_t n) {
    size_t i = (size_t)blockIdx.x * blockDim.x + threadIdx.x;
    size_t stride = (size_t)gridDim.x * blockDim.x;
    for (; i < n; i += stride) p[i] = 0.0f;
}

// ---------------- per-(node,relation) weighted degree ----------------
__global__ void deg_kernel(const int* __restrict__ ei, const int* __restrict__ et,
                           const float* __restrict__ ew, float* __restrict__ degR, int E) {
    int e = blockIdx.x * blockDim.x + threadIdx.x;
    if (e >= E) return;
    int dst = ei[E + e];
    int r   = et[e];
    unsafeAtomicAdd(&degR[(size_t)dst * NRELC + r], ew[e]);
}

// ---------------- per-relation weighted scatter: S[dst,:] += w * x[src,:] ----
__global__ __launch_bounds__(256) void scatter_kernel(
    const float* __restrict__ x, const int* __restrict__ ei,
    const int* __restrict__ et, const float* __restrict__ ew,
    float* __restrict__ S, int E, int rel) {
    int e = blockIdx.x * 8 + (threadIdx.x >> 5);
    if (e >= E) return;
    if (et[e] != rel) return;
    int lane = threadIdx.x & 31;
    float w = ew[e];
    const float* xs = x + (size_t)ei[e] * DIMC;
    float* sd = S + (size_t)ei[E + e] * DIMC;
    #pragma unroll
    for (int i = lane; i < DIMC; i += 32)
        unsafeAtomicAdd(&sd[i], w * xs[i]);
}

// ---------------- fp32 WMMA GEMM with async-LDS double buffering -----------
// D[M,Nout] = A[M,K] @ B[K,Nout] (+epilogue)
// EPI: 0 = store, 1 = accumulate (D += acc), 2 = exact GELU(acc + bias[col])
// Block: 256 thr = 8 waves (4x2); block tile 128x128; wave tile 32x64.
template <int EPI>
__global__ __launch_bounds__(256) void gemm_wmma_kernel(
    const float* __restrict__ A, const float* __restrict__ B,
    float* __restrict__ D, const float* __restrict__ bias,
    int M, int K, int Nout) {
    __shared__ __align__(16) float ldsA[2][BM * LDA_PAD];
    __shared__ __align__(16) float ldsB[2][BK * LDB_PAD];

    const int tid  = threadIdx.x;
    const int lane = tid & 31;
    const int wave = tid >> 5;            // 0..7
    const int wm   = wave & 3;            // 4 M positions of 32 rows
    const int wn   = wave >> 2;           // 2 N positions of 64 cols
    const int half = lane >> 4;           // 0 or 1 (K sub-split)
    const int l16  = lane & 15;

    const int m0b = blockIdx.x * BM;
    const int n0b = blockIdx.y * BN;

    // ---- async staging: 4 b128 chunks of A + 4 of B per thread per K-tile
    // A tile: BM x BK (8 chunks/row), B tile: BK x BN (32 chunks/row)
    int a_l[4], b_l[4];
    #pragma unroll
    for (int i = 0; i < 4; ++i) { a_l[i] = i * 256 + tid; b_l[i] = i * 256 + tid; }

    auto stage = [&](int buf, int k0) {
        #pragma unroll
        for (int i = 0; i < 4; ++i) {
            int row = a_l[i] >> 3;
            int ch  = a_l[i] & 7;
            int gr  = m0b + row; if (gr >= M) gr = M - 1;
            const float* gp = A + (size_t)gr * K + (k0 + ch * 4);
            uint32_t lo = (uint32_t)(uintptr_t)&ldsA[buf][row * LDA_PAD + ch * 4];
            async_copy_b128(lo, gp);
        }
        #pragma unroll
        for (int i = 0; i < 4; ++i) {
            int row = b_l[i] >> 5;
            int ch  = b_l[i] & 31;
            const float* gp = B + (size_t)(k0 + row) * Nout + (n0b + ch * 4);
            uint32_t lo = (uint32_t)(uintptr_t)&ldsB[buf][row * LDB_PAD + ch * 4];
            async_copy_b128(lo, gp);
        }
    };

    v8f acc0[4] = {};
    v8f acc1[4] = {};

    const int r0 = wm * 32 + l16;          // LDS A row for m-tile 0 fragment
    const int c0 = wn * 64 + l16;          // LDS B col base

    const int KT = K / BK;
    stage(0, 0);
    for (int kt = 0; kt < KT; ++kt) {
        const int buf = kt & 1;
        if (kt + 1 < KT) { stage(buf ^ 1, (kt + 1) * BK); wait_async_le8(); }
        else             { wait_async_0(); }
        __syncthreads();

        #pragma unroll
        for (int ks = 0; ks < BK / 4; ++ks) {
            const int kk = ks * 4 + half * 2;   // this half-wave supplies K=kk,kk+1
            v2f a0 = *(const v2f*)&ldsA[buf][r0 * LDA_PAD + kk];
            v2f a1 = *(const v2f*)&ldsA[buf][(r0 + 16) * LDA_PAD + kk];
            v2f bb[4];
            #pragma unroll
            for (int t = 0; t < 4; ++t) {
                bb[t].x = ldsB[buf][kk * LDB_PAD + c0 + t * 16];
                bb[t].y = ldsB[buf][(kk + 1) * LDB_PAD + c0 + t * 16];
            }
            #pragma unroll
            for (int t = 0; t < 4; ++t)
                acc0[t] = __builtin_amdgcn_wmma_f32_16x16x4_f32(
                    false, a0, false, bb[t], (short)0, acc0[t], false, false);
            #pragma unroll
            for (int t = 0; t < 4; ++t)
                acc1[t] = __builtin_amdgcn_wmma_f32_16x16x4_f32(
                    false, a1, false, bb[t], (short)0, acc1[t], false, false);
        }
        __syncthreads();
    }

    // ---- epilogue: C/D layout: VGPR j -> row base+j (lanes 0-15) / base+8+j
    #pragma unroll
    for (int mt = 0; mt < 2; ++mt) {
        const int rbase = m0b + wm * 32 + mt * 16 + half * 8;
        #pragma unroll
        for (int t = 0; t < 4; ++t) {
            const int col = n0b + wn * 64 + t * 16 + l16;
            const float bv = (EPI == 2) ? bias[col] : 0.0f;
            #pragma unroll
            for (int j = 0; j < 8; ++j) {
                const int row = rbase + j;
                if (row < M) {
                    const size_t idx = (size_t)row * Nout + col;
                    float v = (mt == 0) ? acc0[t][j] : acc1[t][j];
                    if (EPI == 2) {
                        v += bv;
                        v = 0.5f * v * (1.0f + erff(v * 0.70710678118f));
                        D[idx] = v;
                    } else if (EPI == 1) {
                        D[idx] += v;
                    } else {
                        D[idx] = v;
                    }
                }
            }
        }
    }
}

// ---------------- LN1: x1 = LN(x + x@Wself + b_self + agg) -----------------
__global__ __launch_bounds__(128) void ln1_kernel(
    const float* __restrict__ x, const float* __restrict__ T1,
    const float* __restrict__ T2, const float* __restrict__ degR,
    const float* __restrict__ b_self, const float* __restrict__ b_rel,
    const float* __restrict__ g, const float* __restrict__ b,
    float* __restrict__ x1) {
    const int n = blockIdx.x;
    const int tid = threadIdx.x;
    __shared__ float s1[128];
    __shared__ float s2[128];

    const float d0 = degR[(size_t)n * 4 + 0];
    const float d1 = degR[(size_t)n * 4 + 1];
    const float d2 = degR[(size_t)n * 4 + 2];
    const float d3 = degR[(size_t)n * 4 + 3];
    const float inv = 1.0f / fmaxf(d0 + d1 + d2 + d3, 1.0f);

    const size_t base = (size_t)n * DIMC;
    float pre[4];
    float lsum = 0.0f, lsq = 0.0f;
    #pragma unroll
    for (int u = 0; u < 4; ++u) {
        const int c = tid + u * 128;
        float bias_msg = d0 * b_rel[c] + d1 * b_rel[DIMC + c] +
                         d2 * b_rel[2 * DIMC + c] + d3 * b_rel[3 * DIMC + c];
        float agg = (T2[base + c] + bias_msg) * inv;
        float p = x[base + c] + T1[base + c] + b_self[c] + agg;
        pre[u] = p;
        lsum += p;
        lsq  += p * p;
    }
    s1[tid] = lsum; s2[tid] = lsq;
    __syncthreads();
    for (int s = 64; s > 0; s >>= 1) {
        if (tid < s) { s1[tid] += s1[tid + s]; s2[tid] += s2[tid + s]; }
        __syncthreads();
    }
    const float mu   = s1[0] * (1.0f / DIMC);
    const float var  = s2[0] * (1.0f / DIMC) - mu * mu;
    const float rstd = rsqrtf(var + 1e-5f);
    #pragma unroll
    for (int u = 0; u < 4; ++u) {
        const int c = tid + u * 128;
        x1[base + c] = (pre[u] - mu) * rstd * g[c] + b[c];
    }
}

// ---------------- LN2: out = LN(x1 + F + b_ffn2) ---------------------------
__global__ __launch_bounds__(128) void ln2_kernel(
    const float* __restrict__ x1, const float* __restrict__ F,
    const float* __restrict__ b2, const float* __restrict__ g,
    const float* __restrict__ b, float* __restrict__ out) {
    const int n = blockIdx.x;
    const int tid = threadIdx.x;
    __shared__ float s1[128];
    __shared__ float s2[128];

    const size_t base = (size_t)n * DIMC;
    float pre[4];
    float lsum = 0.0f, lsq = 0.0f;
    #pragma unroll
    for (int u = 0; u < 4; ++u) {
        const int c = tid + u * 128;
        float p = x1[base + c] + F[base + c] + b2[c];
        pre[u] = p;
        lsum += p;
        lsq  += p * p;
    }
    s1[tid] = lsum; s2[tid] = lsq;
    __syncthreads();
    for (int s = 64; s > 0; s >>= 1) {
        if (tid < s) { s1[tid] += s1[tid + s]; s2[tid] += s2[tid + s]; }
        __syncthreads();
    }
    const float mu   = s1[0] * (1.0f / DIMC);
    const float var  = s2[0] * (1.0f / DIMC) - mu * mu;
    const float rstd = rsqrtf(var + 1e-5f);
    #pragma unroll
    for (int u = 0; u < 4; ++u) {
        const int c = tid + u * 128;
        out[base + c] = (pre[u] - mu) * rstd * g[c] + b[c];
    }
}

extern "C" void kernel_launch(void* const* d_in, const int* in_sizes, int n_in,
                              void* d_out, int out_size, void* d_ws, size_t ws_size,
                              hipStream_t stream) {
    const float* x      = (const float*)d_in[0];
    const int*   ei     = (const int*)d_in[1];
    const int*   et     = (const int*)d_in[2];
    const float* ew     = (const float*)d_in[3];
    const float* W_self = (const float*)d_in[4];
    const float* b_self = (const float*)d_in[5];
    const float* W_rel  = (const float*)d_in[6];
    const float* b_rel  = (const float*)d_in[7];
    const float* ln1g   = (const float*)d_in[8];
    const float* ln1b   = (const float*)d_in[9];
    const float* ln2g   = (const float*)d_in[10];
    const float* ln2b   = (const float*)d_in[11];
    const float* W_ffn1 = (const float*)d_in[12];
    const float* b_ffn1 = (const float*)d_in[13];
    const float* W_ffn2 = (const float*)d_in[14];
    const float* b_ffn2 = (const float*)d_in[15];
    float* out = (float*)d_out;

    const int N   = in_sizes[0] / DIMC;
    const int E   = in_sizes[2];
    const int HID = in_sizes[13]; // 1024

    // workspace (floats): [S | T1 | T2 | X1 | degR]; G1 aliases S+T1, F aliases T2
    float* ws = (float*)d_ws;
    const size_t NS = (size_t)N * DIMC;
    float* S    = ws;
    float* T1   = ws + NS;
    float* T2   = ws + 2 * NS;
    float* X1   = ws + 3 * NS;
    float* degR = ws + 4 * NS;
    float* G1   = ws;       // N x HID, reuses S+T1 after consumed
    float* F    = T2;       // N x DIM, reuses T2 after LN1

    const dim3 blk(256);
    const dim3 gD((unsigned)((N + BM - 1) / BM), (unsigned)(DIMC / BN));
    const dim3 gH((unsigned)((N + BM - 1) / BM), (unsigned)(HID / BN));

    // degree accumulation
    zero_kernel<<<256, blk, 0, stream>>>(degR, (size_t)N * NRELC);
    deg_kernel<<<(E + 255) / 256, blk, 0, stream>>>(ei, et, ew, degR, E);

    // T1 = x @ W_self
    gemm_wmma_kernel<0><<<gD, blk, 0, stream>>>(x, W_self, T1, nullptr, N, DIMC, DIMC);

    // T2 = sum_r S_r @ W_rel[r]
    for (int r = 0; r < NRELC; ++r) {
        zero_kernel<<<2048, blk, 0, stream>>>(S, NS);
        scatter_kernel<<<(E + 7) / 8, blk, 0, stream>>>(x, ei, et, ew, S, E, r);
        if (r == 0)
            gemm_wmma_kernel<0><<<gD, blk, 0, stream>>>(S, W_rel + (size_t)r * DIMC * DIMC,
                                                        T2, nullptr, N, DIMC, DIMC);
        else
            gemm_wmma_kernel<1><<<gD, blk, 0, stream>>>(S, W_rel + (size_t)r * DIMC * DIMC,
                                                        T2, nullptr, N, DIMC, DIMC);
    }

    // X1 = LN1(x + T1 + b_self + normalized agg)
    ln1_kernel<<<N, 128, 0, stream>>>(x, T1, T2, degR, b_self, b_rel, ln1g, ln1b, X1);

    // G1 = gelu(X1 @ W_ffn1 + b_ffn1)   [N, HID]
    gemm_wmma_kernel<2><<<gH, blk, 0, stream>>>(X1, W_ffn1, G1, b_ffn1, N, DIMC, HID);

    // F = G1 @ W_ffn2                   [N, DIM]
    gemm_wmma_kernel<0><<<gD, blk, 0, stream>>>(G1, W_ffn2, F, nullptr, N, HID, DIMC);

    // out = LN2(X1 + F + b_ffn2)
    ln2_kernel<<<N, 128, 0, stream>>>(X1, F, b_ffn2, ln2g, ln2b, out);
}